// Attention_76287209112099
// MI455X (gfx1250) — compile-verified
//
#include <hip/hip_runtime.h>
#include <math.h>
#include <stdint.h>

// ---------------------------------------------------------------------------
// CDNA5 WMMA f32 16x16x4:  A = 16x4 f32 (2 VGPR), B = 4x16 f32 (2 VGPR),
// C/D = 16x16 f32 (8 VGPR).
// Fragment layouts (ISA 7.12.2):
//   A: lane L -> m = L%16 ; v0 = A[m, k0 + (L>=16?2:0)], v1 = A[m, k0 + (L>=16?3:1)]
//   B: lane L -> n = L%16 ; v0 = B[k0 + (L>=16?2:0), n], v1 = B[k0 + (L>=16?3:1), n]
//   C/D: VGPR r, lane L -> row = r + (L>=16?8:0), col = L%16
// Every GEMM here is K-contiguous on both operands, so A/B fragments are
// plain per-lane float2 loads.
// ---------------------------------------------------------------------------
typedef float v2f __attribute__((ext_vector_type(2)));
typedef float v8f __attribute__((ext_vector_type(8)));

__device__ __forceinline__ v8f wmma4(v2f a, v2f b, v8f c) {
  return __builtin_amdgcn_wmma_f32_16x16x4_f32(false, a, false, b, (short)0, c,
                                               false, false);
}

__device__ __forceinline__ float xmax16(float v) {
  v = fmaxf(v, __shfl_xor(v, 1));
  v = fmaxf(v, __shfl_xor(v, 2));
  v = fmaxf(v, __shfl_xor(v, 4));
  v = fmaxf(v, __shfl_xor(v, 8));
  return v;
}
__device__ __forceinline__ float xsum16(float v) {
  v += __shfl_xor(v, 1);
  v += __shfl_xor(v, 2);
  v += __shfl_xor(v, 4);
  v += __shfl_xor(v, 8);
  return v;
}

#define BATCH 2
#define SEQ   2048
#define DMODEL 1024
#define NHEAD 16
#define DHEAD 64
#define MROWS (BATCH * SEQ)      // 4096
#define NQKV  (3 * DMODEL)       // 3072

// ---------------------------------------------------------------------------
// Kernel 1: qkv = x @ Wqkv^T, scattered into head-major q/k/v [B,H,S,DHEAD]
// 256 thr = 8 waves (4Mx2N), block tile 128x128, wave tile 32x64.
// ---------------------------------------------------------------------------
__global__ __launch_bounds__(256) void gemm_qkv(
    const float* __restrict__ X, const float* __restrict__ W,
    float* __restrict__ qb, float* __restrict__ kb, float* __restrict__ vb) {
  const int K = DMODEL;
  const int lane = threadIdx.x & 31;
  const int wave = threadIdx.x >> 5;
  const int col = lane & 15;
  const int half = lane >> 4;
  const int koff = half * 2;
  const int m0 = blockIdx.y * 128 + (wave >> 1) * 32;
  const int n0 = blockIdx.x * 128 + (wave & 1) * 64;

  v8f acc[2][4] = {};

  const float* a0 = X + (size_t)(m0 + col) * K + koff;
  const float* a1 = a0 + (size_t)16 * K;
  const float* b0p = W + (size_t)(n0 + col) * K + koff;
  const float* b1p = b0p + (size_t)16 * K;
  const float* b2p = b0p + (size_t)32 * K;
  const float* b3p = b0p + (size_t)48 * K;

#pragma unroll 4
  for (int k = 0; k < K; k += 4) {
    v2f av0 = *(const v2f*)(a0 + k);
    v2f av1 = *(const v2f*)(a1 + k);
    v2f bv0 = *(const v2f*)(b0p + k);
    v2f bv1 = *(const v2f*)(b1p + k);
    v2f bv2 = *(const v2f*)(b2p + k);
    v2f bv3 = *(const v2f*)(b3p + k);
    acc[0][0] = wmma4(av0, bv0, acc[0][0]);
    acc[0][1] = wmma4(av0, bv1, acc[0][1]);
    acc[0][2] = wmma4(av0, bv2, acc[0][2]);
    acc[0][3] = wmma4(av0, bv3, acc[0][3]);
    acc[1][0] = wmma4(av1, bv0, acc[1][0]);
    acc[1][1] = wmma4(av1, bv1, acc[1][1]);
    acc[1][2] = wmma4(av1, bv2, acc[1][2]);
    acc[1][3] = wmma4(av1, bv3, acc[1][3]);
  }

#pragma unroll
  for (int i = 0; i < 2; ++i) {
#pragma unroll
    for (int j = 0; j < 4; ++j) {
      const int nbase = n0 + j * 16;
      const int which = nbase >> 10;  // uniform per accumulator
      const int rem = nbase & 1023;
      const int h = rem >> 6;
      const int dh0 = rem & 63;
      float* dst = (which == 0) ? qb : ((which == 1) ? kb : vb);
#pragma unroll
      for (int r = 0; r < 8; ++r) {
        const int m = m0 + i * 16 + r + half * 8;
        const int b = m >> 11;
        const int s = m & 2047;
        dst[((size_t)((b << 4) | h) * SEQ + s) * DHEAD + dh0 + col] =
            acc[i][j][r];
      }
    }
  }
}

// ---------------------------------------------------------------------------
// Kernel 2: in-place RoPE on q and k, layout [B*H, S, DHEAD], pos = s.
// ---------------------------------------------------------------------------
__global__ __launch_bounds__(256) void rope_kernel(float* __restrict__ qbuf,
                                                   float* __restrict__ kbuf) {
  const size_t per = (size_t)BATCH * NHEAD * SEQ * (DHEAD / 2);  // 2,097,152
  size_t idx = (size_t)blockIdx.x * blockDim.x + threadIdx.x;
  float* buf = qbuf;
  if (idx >= per) { buf = kbuf; idx -= per; }
  const int pair = (int)(idx & 31);
  const int s = (int)((idx >> 5) & (SEQ - 1));
  const float inv_freq = powf(10000.0f, -((float)(2 * pair) / (float)DHEAD));
  float c, sn;
  sincosf((float)s * inv_freq, &sn, &c);
  float* p = buf + (idx >> 5) * DHEAD + pair * 2;
  const float x1 = p[0];
  const float x2 = p[1];
  p[0] = x1 * c - x2 * sn;
  p[1] = x1 * sn + x2 * c;
}

// ---------------------------------------------------------------------------
// Kernel 3: causal flash attention per (b,h). 128 thr = 4 waves, each wave
// owns a 16-row q tile; streams 16-key tiles with online softmax.
// K/V tiles staged in LDS via ASYNC global->LDS copies (double-buffered K,
// prefetched V) and consumed with WMMA. grid.x = B*H*(S/64) = 1024.
//
// K tile is stored chunk-XOR swizzled: element (row r, dim x) lives at
// byte  r*256 + ((x>>2) ^ r)*16 + (x&3)*4   -> the QK^T B-fragment read
// (lane=col reads row col, dims 4d..4d+3) hits 16 distinct LDS banks.
// V tile is linear [16][64].
// ---------------------------------------------------------------------------
__global__ __launch_bounds__(128) void flash_attn(
    const float* __restrict__ qbuf, const float* __restrict__ kbuf,
    const float* __restrict__ vbuf, float* __restrict__ obuf) {
  __shared__ __align__(16) float skbuf[4][2][16 * 64];  // swizzled K, 2 slots
  __shared__ __align__(16) float svbuf[4][16 * 64];     // linear V, 1 slot
  __shared__ __align__(16) float plds[4][16][18];       // P tile (padded)

  const int lane = threadIdx.x & 31;
  const int wave = threadIdx.x >> 5;
  const int col = lane & 15;
  const int half = lane >> 4;
  const int koff = half * 2;

  const int bh = blockIdx.x >> 5;  // 0..31  (B*H)
  const int qt = blockIdx.x & 31;  // q tile of 64
  const int q0 = qt * 64 + wave * 16;

  const size_t hb = (size_t)bh * SEQ * DHEAD;
  const float* qh = qbuf + hb;
  const float* kh = kbuf + hb;
  const float* vh = vbuf + hb;

  // async-copy lane mapping: lane moves one 16B chunk (cc) of row 2t+rp
  const int cc = col;
  const int rp = half;
  const uint32_t klds[2] = {(uint32_t)(uintptr_t)&skbuf[wave][0][0],
                            (uint32_t)(uintptr_t)&skbuf[wave][1][0]};
  const uint32_t vlds = (uint32_t)(uintptr_t)&svbuf[wave][0];

// 8 x b128 per tile: 32 lanes x 16B = 512B/instr, 4KB tile.
#define PREFETCH_K(KLDS, K0)                                                 \
  {                                                                          \
    _Pragma("unroll") for (int t = 0; t < 8; ++t) {                          \
      const int r = 2 * t + rp;                                              \
      uint32_t ka = (KLDS) + (uint32_t)(r * 256 + ((cc ^ r) << 4));          \
      uint32_t go = (uint32_t)(((K0) + r) * 256 + cc * 16);                  \
      asm volatile("global_load_async_to_lds_b128 %0, %1, %2" ::"v"(ka),     \
                   "v"(go), "s"(kh)                                          \
                   : "memory");                                              \
    }                                                                        \
  }
#define PREFETCH_V(K0)                                                       \
  {                                                                          \
    _Pragma("unroll") for (int t = 0; t < 8; ++t) {                          \
      const int r = 2 * t + rp;                                              \
      uint32_t va = vlds + (uint32_t)(r * 256 + cc * 16);                    \
      uint32_t go = (uint32_t)(((K0) + r) * 256 + cc * 16);                  \
      asm volatile("global_load_async_to_lds_b128 %0, %1, %2" ::"v"(va),     \
                   "v"(go), "s"(vh)                                          \
                   : "memory");                                              \
    }                                                                        \
  }

  // Kick off K tile 0 while we fetch q.
  PREFETCH_K(klds[0], 0);

  // Preload q fragments (16 x v2f = full 16x64 tile striped over the wave)
  v2f qf[16];
  {
    const float* qrow = qh + (size_t)(q0 + col) * DHEAD + koff;
#pragma unroll
    for (int d = 0; d < 16; ++d) qf[d] = *(const v2f*)(qrow + d * 4);
  }

  v8f oacc[4] = {};
  float mrun[8], lrun[8];
#pragma unroll
  for (int r = 0; r < 8; ++r) { mrun[r] = -1e30f; lrun[r] = 0.0f; }

  const int nkt = (q0 >> 4) + 1;  // causal: tiles 0 .. q0/16
  for (int kt = 0; kt < nkt; ++kt) {
    const int cur = kt & 1;
    const int k0 = kt << 4;
    const bool more = (kt + 1 < nkt);

    // Outstanding at entry: K(cur)=8.  Issue V(cur), then K(next).
    PREFETCH_V(k0);
    if (more) {
      PREFETCH_K(klds[cur ^ 1], k0 + 16);
      asm volatile("s_wait_asynccnt 0x10" ::: "memory");  // K(cur) done
    } else {
      asm volatile("s_wait_asynccnt 0x8" ::: "memory");   // K(cur) done
    }

    // S = q @ k^T over DHEAD; B-fragments from swizzled LDS K tile.
    const float* kl = &skbuf[wave][cur][0];
    v8f s = {};
#pragma unroll
    for (int d = 0; d < 16; ++d) {
      v2f kf = *(const v2f*)&kl[col * 64 + ((d ^ col) << 2) + koff];
      s = wmma4(qf[d], kf, s);
    }

    const bool diag = (k0 == q0);  // only the last tile straddles the diagonal

    // Online softmax (row = r + 8*half lives across 16 lanes of one VGPR)
#pragma unroll
    for (int r = 0; r < 8; ++r) {
      float sv = s[r] * 0.125f;  // 1/sqrt(64)
      if (diag && (k0 + col > q0 + r + half * 8)) sv = -1e30f;
      const float rowmax = xmax16(sv);
      const float mnew = fmaxf(mrun[r], rowmax);
      const float corr = __expf(mrun[r] - mnew);
      const float p = __expf(sv - mnew);
      lrun[r] = lrun[r] * corr + xsum16(p);
      mrun[r] = mnew;
#pragma unroll
      for (int dt = 0; dt < 4; ++dt) oacc[dt][r] *= corr;
      plds[wave][r + half * 8][col] = p;  // C-layout -> LDS
    }
    asm volatile("s_wait_dscnt 0" ::: "memory");  // P tile visible in-wave

    // Wait for V(cur); only K(next) may stay outstanding.
    if (more) asm volatile("s_wait_asynccnt 0x8" ::: "memory");
    else      asm volatile("s_wait_asynccnt 0x0" ::: "memory");

    // O += P @ V  (A = P from LDS, B = V rows from LDS)
    const float* vl = &svbuf[wave][0];
#pragma unroll
    for (int dt = 0; dt < 4; ++dt) {
      const int dcol = dt * 16 + col;
#pragma unroll
      for (int kk = 0; kk < 16; kk += 4) {
        v2f a = *(const v2f*)&plds[wave][col][kk + koff];
        v2f b;
        b.x = vl[(kk + koff) * 64 + dcol];
        b.y = vl[(kk + koff + 1) * 64 + dcol];
        oacc[dt] = wmma4(a, b, oacc[dt]);
      }
    }
  }
#undef PREFETCH_K
#undef PREFETCH_V

  // Store O in [B*S, DMODEL] so the output projection can consume it as A.
  const int b = bh >> 4;
  const int h = bh & 15;
  float* orow = obuf + (size_t)b * SEQ * DMODEL + (size_t)h * DHEAD;
#pragma unroll
  for (int dt = 0; dt < 4; ++dt) {
#pragma unroll
    for (int r = 0; r < 8; ++r) {
      const int m = q0 + r + half * 8;
      orow[(size_t)m * DMODEL + dt * 16 + col] = oacc[dt][r] / lrun[r];
    }
  }
}

// ---------------------------------------------------------------------------
// Kernel 4: out = o @ Wo^T  (M=4096, N=1024, K=1024), row-major store.
// ---------------------------------------------------------------------------
__global__ __launch_bounds__(256) void gemm_out(const float* __restrict__ A,
                                                const float* __restrict__ W,
                                                float* __restrict__ C) {
  const int K = DMODEL;
  const int lane = threadIdx.x & 31;
  const int wave = threadIdx.x >> 5;
  const int col = lane & 15;
  const int half = lane >> 4;
  const int koff = half * 2;
  const int m0 = blockIdx.y * 128 + (wave >> 1) * 32;
  const int n0 = blockIdx.x * 128 + (wave & 1) * 64;

  v8f acc[2][4] = {};

  const float* a0 = A + (size_t)(m0 + col) * K + koff;
  const float* a1 = a0 + (size_t)16 * K;
  const float* b0p = W + (size_t)(n0 + col) * K + koff;
  const float* b1p = b0p + (size_t)16 * K;
  const float* b2p = b0p + (size_t)32 * K;
  const float* b3p = b0p + (size_t)48 * K;

#pragma unroll 4
  for (int k = 0; k < K; k += 4) {
    v2f av0 = *(const v2f*)(a0 + k);
    v2f av1 = *(const v2f*)(a1 + k);
    v2f bv0 = *(const v2f*)(b0p + k);
    v2f bv1 = *(const v2f*)(b1p + k);
    v2f bv2 = *(const v2f*)(b2p + k);
    v2f bv3 = *(const v2f*)(b3p + k);
    acc[0][0] = wmma4(av0, bv0, acc[0][0]);
    acc[0][1] = wmma4(av0, bv1, acc[0][1]);
    acc[0][2] = wmma4(av0, bv2, acc[0][2]);
    acc[0][3] = wmma4(av0, bv3, acc[0][3]);
    acc[1][0] = wmma4(av1, bv0, acc[1][0]);
    acc[1][1] = wmma4(av1, bv1, acc[1][1]);
    acc[1][2] = wmma4(av1, bv2, acc[1][2]);
    acc[1][3] = wmma4(av1, bv3, acc[1][3]);
  }

#pragma unroll
  for (int i = 0; i < 2; ++i)
#pragma unroll
    for (int j = 0; j < 4; ++j)
#pragma unroll
      for (int r = 0; r < 8; ++r) {
        const int m = m0 + i * 16 + r + half * 8;
        const int n = n0 + j * 16 + col;
        C[(size_t)m * DMODEL + n] = acc[i][j][r];
      }
}

// ---------------------------------------------------------------------------
// Host launcher
// ---------------------------------------------------------------------------
extern "C" void kernel_launch(void* const* d_in, const int* in_sizes, int n_in,
                              void* d_out, int out_size, void* d_ws,
                              size_t ws_size, hipStream_t stream) {
  (void)in_sizes; (void)n_in; (void)out_size; (void)ws_size;
  const float* x = (const float*)d_in[0];     // [B,S,D]
  const float* Wqkv = (const float*)d_in[1];  // [3D,D]
  const float* Wo = (const float*)d_in[2];    // [D,D]
  float* out = (float*)d_out;                 // [B,S,D]

  float* ws = (float*)d_ws;
  const size_t HEADSZ = (size_t)BATCH * NHEAD * SEQ * DHEAD;  // 4,194,304
  float* qb = ws;
  float* kb = ws + HEADSZ;
  float* vb = ws + 2 * HEADSZ;
  float* ob = ws + 3 * HEADSZ;  // [B*S, DMODEL]

  gemm_qkv<<<dim3(NQKV / 128, MROWS / 128), 256, 0, stream>>>(x, Wqkv, qb, kb,
                                                              vb);
  {
    const size_t total = 2 * (size_t)BATCH * NHEAD * SEQ * (DHEAD / 2);
    rope_kernel<<<(unsigned)(total / 256), 256, 0, stream>>>(qb, kb);
  }
  flash_attn<<<BATCH * NHEAD * (SEQ / 64), 128, 0, stream>>>(qb, kb, vb, ob);
  gemm_out<<<dim3(DMODEL / 128, MROWS / 128), 256, 0, stream>>>(ob, Wo, out);
}